// MulitAttention_17987323036106
// MI455X (gfx1250) — compile-verified
//
#include <hip/hip_runtime.h>

// MI455X / gfx1250, wave32. f32 WMMA (16x16x4) for both GEMMs (HBM-bound:
// 1.6 GFLOP vs 201 MB f32 input -> keep f32 fidelity; matrix pipes do MACs).
// Tiles stream global->LDS via CDNA5 async ops (ASYNCcnt), double-buffered so
// the async engine overlaps the WMMA loop. A-matrices are zero-padded to 16
// rows in LDS so fragment loads are unconditional (no EXEC toggling).

typedef float v2f __attribute__((ext_vector_type(2)));
typedef float v8f __attribute__((ext_vector_type(8)));
typedef float f4  __attribute__((ext_vector_type(4)));

#define B_  16
#define S_  4096
#define H_  768
#define E_  8
#define EP_ 16            // heads padded to WMMA M=16

#define SW_STRIDE  772    // ==4 mod 64: 16 rows distinct banks; even -> b64-aligned
#define SI1_STRIDE 36     // 144B rows: 16B aligned; 36i mod 64 multiple of 4 -> conflict-free
#define SI3_STRIDE 136    // 544B rows: 16B aligned; 2*136 mod 64 == 16 -> halves disjoint
#define SP_STRIDE  34     // even -> aligned pairs; 34D==2 mod 64 unsolvable -> conflict-free

#define TILE1 (128 * SI1_STRIDE)
#define TILE3 (32 * SI3_STRIDE)

__device__ __forceinline__ void async_ld_b128(unsigned lds_addr, const void* gaddr) {
    asm volatile("global_load_async_to_lds_b128 %0, %1, off"
                 :: "v"(lds_addr), "v"((unsigned long long)(uintptr_t)gaddr)
                 : "memory");
}
__device__ __forceinline__ void wait_async0() {
    asm volatile("s_wait_asynccnt 0x0" ::: "memory");
}

// ---------------------------------------------------------------------------
// Kernel 1: atten[b,e,s] = sum_h W[e,h]*x[b,s,h] + bias[e]
// One block = (batch, 128-seq tile). 8 waves, each owns a 16x16 head x seq tile.
// ---------------------------------------------------------------------------
__global__ __launch_bounds__(256) void k_logits(const float* __restrict__ inp,
                                                const float* __restrict__ W,
                                                const float* __restrict__ bias,
                                                float* __restrict__ atten) {
    __shared__ float sW[EP_ * SW_STRIDE];      // 48.3 KB, rows 8..15 zero
    __shared__ float sIn[2 * TILE1];           // 36 KB double-buffered

    const int tid   = threadIdx.x;
    const int lane  = tid & 31;
    const int wv    = tid >> 5;            // 0..7
    const int bb    = blockIdx.x >> 5;     // batch
    const int sTile = (blockIdx.x & 31) * 128;

    const int nm = lane & 15;              // A row (head) / B col (seq) index
    const int hi = lane >> 4;              // K-half selector

    // stage W zero-padded to 16 rows (float4, rows 3088B -> 16B aligned)
    for (int i = tid; i < EP_ * (H_ / 4); i += 256) {
        const int e = i / (H_ / 4), h4 = i % (H_ / 4);
        f4 v = {0.f, 0.f, 0.f, 0.f};
        if (e < E_) v = *(const f4*)&W[e * H_ + h4 * 4];
        *(f4*)&sW[e * SW_STRIDE + h4 * 4] = v;
    }

    const float* inBase = inp + (size_t)bb * S_ * H_;
    const unsigned sInBase = (unsigned)(uintptr_t)&sIn[0];
    v8f c = {};

    // async fill of one 128x32 tile: 1024 x b128, 4 per thread
    auto issue = [&](int t) {
        const unsigned bufB = sInBase + (unsigned)((t & 1) * TILE1) * 4u;
        const int k0 = t * 32;
        #pragma unroll
        for (int j = 0; j < 4; ++j) {
            const int idx = tid + j * 256;
            const int r = idx >> 3, q = idx & 7;           // 8 float4 per row
            async_ld_b128(bufB + (unsigned)(r * SI1_STRIDE + q * 4) * 4u,
                          inBase + (size_t)(sTile + r) * H_ + k0 + q * 4);
        }
    };

    issue(0);
    for (int t = 0; t < H_ / 32; ++t) {
        wait_async0();                      // tile t resident (this wave's issues)
        __syncthreads();                    // all waves resident + tile t-1 consumed
        if (t + 1 < H_ / 32) issue(t + 1);  // stream next tile during compute
        const float* buf = &sIn[(t & 1) * TILE1];
        const int k0 = t * 32;
        #pragma unroll
        for (int kk = 0; kk < 32; kk += 4) {
            v2f a = { sW[nm * SW_STRIDE + k0 + kk + 2 * hi],
                      sW[nm * SW_STRIDE + k0 + kk + 1 + 2 * hi] };
            const int sr = wv * 16 + nm;
            v2f b = { buf[sr * SI1_STRIDE + kk + 2 * hi],
                      buf[sr * SI1_STRIDE + kk + 1 + 2 * hi] };
            c = __builtin_amdgcn_wmma_f32_16x16x4_f32(
                    false, a, false, b, (short)0, c, false, false);
        }
    }

    if (lane < 16) {                        // lanes 0..15 hold heads 0..7 (VGPR r)
        const int s = sTile + wv * 16 + lane;
        #pragma unroll
        for (int r = 0; r < E_; ++r)
            atten[((size_t)bb * E_ + r) * S_ + s] = c[r] + bias[r];
    }
}

// ---------------------------------------------------------------------------
// Kernel 2: row softmax over S=4096 for each of the 128 (b,e) rows.
// ---------------------------------------------------------------------------
__global__ __launch_bounds__(256) void k_softmax(const float* __restrict__ atten,
                                                 float* __restrict__ p) {
    __shared__ float red[256];
    const int row = blockIdx.x;
    const int tid = threadIdx.x;
    const float* x = atten + (size_t)row * S_;

    float mx = -3.402823466e38f;
    for (int s = tid; s < S_; s += 256) mx = fmaxf(mx, x[s]);
    red[tid] = mx; __syncthreads();
    for (int off = 128; off > 0; off >>= 1) {
        if (tid < off) red[tid] = fmaxf(red[tid], red[tid + off]);
        __syncthreads();
    }
    mx = red[0]; __syncthreads();

    float sum = 0.f;
    for (int s = tid; s < S_; s += 256) sum += expf(x[s] - mx);
    red[tid] = sum; __syncthreads();
    for (int off = 128; off > 0; off >>= 1) {
        if (tid < off) red[tid] += red[tid + off];
        __syncthreads();
    }
    const float inv = 1.f / red[0];

    float* pr = p + (size_t)row * S_;
    for (int s = tid; s < S_; s += 256) pr[s] = expf(x[s] - mx) * inv;
}

// ---------------------------------------------------------------------------
// Kernel 3: out[b,e,h] = sum_s p[b,e,s]*x[b,s,h]; K split 4 ways into partials.
// grid = (B, 6 n-tiles of 128 cols, 4 k-chunks of 1024). 8 waves / block.
// ---------------------------------------------------------------------------
__global__ __launch_bounds__(256) void k_out(const float* __restrict__ inp,
                                             const float* __restrict__ p,
                                             float* __restrict__ partial) {
    __shared__ float sIn[2 * TILE3];           // 34 KB double-buffered
    __shared__ float sP[2 * EP_ * SP_STRIDE];  // 4.3 KB, rows 8..15 zero

    const int tid  = threadIdx.x;
    const int lane = tid & 31;
    const int wv   = tid >> 5;
    const int bb   = blockIdx.x;
    const int nt   = blockIdx.y;               // 0..5
    const int kc   = blockIdx.z;               // 0..3
    const int nm   = lane & 15;
    const int hi   = lane >> 4;

    const float* inBase = inp + (size_t)bb * S_ * H_;
    const float* pBase  = p   + (size_t)bb * E_ * S_;
    const unsigned sInBase = (unsigned)(uintptr_t)&sIn[0];
    v8f c = {};

    // zero the padding rows (8..15) of both sP buffers once
    for (int i = tid; i < 2 * E_ * SP_STRIDE; i += 256) {
        const int bufi = i / (E_ * SP_STRIDE), r = i % (E_ * SP_STRIDE);
        sP[bufi * EP_ * SP_STRIDE + E_ * SP_STRIDE + r] = 0.f;
    }

    auto issue = [&](int t) {                  // tile t: 32 seq rows starting s0
        const int s0 = kc * 1024 + t * 32;
        const unsigned bufB = sInBase + (unsigned)((t & 1) * TILE3) * 4u;
        #pragma unroll
        for (int j = 0; j < 4; ++j) {
            const int idx = tid + j * 256;
            const int r = idx >> 5, q = idx & 31;          // 32 float4 per row
            async_ld_b128(bufB + (unsigned)(r * SI3_STRIDE + q * 4) * 4u,
                          inBase + (size_t)(s0 + r) * H_ + nt * 128 + q * 4);
        }
        // probability tile 8x32: exactly one float per thread (e=tid/32)
        const int e = tid >> 5, so = tid & 31;
        sP[(t & 1) * EP_ * SP_STRIDE + e * SP_STRIDE + so] =
            pBase[(size_t)e * S_ + s0 + so];
    };

    issue(0);
    for (int t = 0; t < 32; ++t) {
        wait_async0();
        __syncthreads();
        if (t + 1 < 32) issue(t + 1);
        const float* bufI = &sIn[(t & 1) * TILE3];
        const float* bufP = &sP[(t & 1) * EP_ * SP_STRIDE];
        #pragma unroll
        for (int ss = 0; ss < 32; ss += 4) {
            v2f a = { bufP[nm * SP_STRIDE + ss + 2 * hi],
                      bufP[nm * SP_STRIDE + ss + 1 + 2 * hi] };
            v2f b = { bufI[(ss     + 2 * hi) * SI3_STRIDE + wv * 16 + nm],
                      bufI[(ss + 1 + 2 * hi) * SI3_STRIDE + wv * 16 + nm] };
            c = __builtin_amdgcn_wmma_f32_16x16x4_f32(
                    false, a, false, b, (short)0, c, false, false);
        }
    }

    if (lane < 16) {
        const int col = nt * 128 + wv * 16 + lane;
        #pragma unroll
        for (int r = 0; r < E_; ++r)
            partial[(((size_t)bb * 4 + kc) * E_ + r) * H_ + col] = c[r];
    }
}

// ---------------------------------------------------------------------------
// Kernel 4: deterministic reduction of the 4 K-chunk partials into d_out.
// ---------------------------------------------------------------------------
__global__ __launch_bounds__(256) void k_reduce(const float* __restrict__ partial,
                                                float* __restrict__ out) {
    const int idx = blockIdx.x * 256 + threadIdx.x;   // over B*E*H
    if (idx >= B_ * E_ * H_) return;
    const int b   = idx / (E_ * H_);
    const int rem = idx - b * (E_ * H_);
    float v = 0.f;
    #pragma unroll
    for (int kc = 0; kc < 4; ++kc)
        v += partial[((size_t)(b * 4 + kc)) * (E_ * H_) + rem];
    out[idx] = v;
}

extern "C" void kernel_launch(void* const* d_in, const int* in_sizes, int n_in,
                              void* d_out, int out_size, void* d_ws, size_t ws_size,
                              hipStream_t stream) {
    const float* inp  = (const float*)d_in[0];   // [16,4096,768]
    const float* W    = (const float*)d_in[1];   // [8,768]
    const float* bias = (const float*)d_in[2];   // [8]

    float* out   = (float*)d_out;                // [16,8,768]
    float* atten = out + (size_t)B_ * E_ * H_;   // [16,8,4096] (pre-softmax logits)

    float* p       = (float*)d_ws;               // [16,8,4096] probabilities (2 MB)
    float* partial = p + (size_t)B_ * E_ * S_;   // [16,4,8,768] K-chunk partials (1.5 MB)

    k_logits <<<dim3(B_ * (S_ / 128)), 256, 0, stream>>>(inp, W, bias, atten);
    k_softmax<<<dim3(B_ * E_),          256, 0, stream>>>(atten, p);
    k_out    <<<dim3(B_, H_ / 128, 4),  256, 0, stream>>>(inp, p, partial);
    k_reduce <<<dim3((B_ * E_ * H_ + 255) / 256), 256, 0, stream>>>(partial, out);
}